// MeanAggregator_2740189135076
// MI455X (gfx1250) — compile-verified
//
#include <hip/hip_runtime.h>

// MeanAggregator on gfx1250 (MI455X).
// out[p, :] = (sum_{n < len[p]} X[p, n, :]) * (1/D)   with P = B*V, D = 128, MAX_N = 64.
// Bandwidth-bound (only ~252 MiB of live rows after length masking -> ~11 us
// floor at 23.3 TB/s). CDNA5 path: async global->LDS copies (ASYNCcnt) with a
// two-chunk software pipeline, LDS reduce, non-temporal I/O.

#define WAVES_PER_BLOCK 4
#define MAXN 64
#define DIM 128        // floats per row; 512 bytes
#define CPOL_NT 1      // TH = non-temporal (stream-once data)

typedef float v4f __attribute__((ext_vector_type(4)));
typedef int   v4i __attribute__((ext_vector_type(4)));

#define AS1 __attribute__((address_space(1)))
#define AS3 __attribute__((address_space(3)))

#if defined(__has_builtin)
#  if __has_builtin(__builtin_amdgcn_global_load_async_to_lds_b128)
#    define HAVE_ASYNC_LDS 1
#  endif
#  if __has_builtin(__builtin_amdgcn_s_wait_asynccnt)
#    define HAVE_WAIT_ASYNC 1
#  endif
#endif

__device__ __forceinline__ void wait_async_zero() {
    asm volatile("" ::: "memory");       // keep LDS reads on the far side
#if defined(HAVE_WAIT_ASYNC)
    __builtin_amdgcn_s_wait_asynccnt(0);
#else
    asm volatile("s_wait_asynccnt 0" ::: "memory");
#endif
    asm volatile("" ::: "memory");
}

#if defined(HAVE_ASYNC_LDS)
// Issue async row copies for rows [a, b). The instruction's 24-bit immediate
// offset is added to BOTH the global and LDS address (row stride 512 B on both
// sides), so a 4x unrolled body reuses one address pair with imm offsets.
__device__ __forceinline__ void issue_rows(const float* __restrict__ src,
                                           float* __restrict__ ldsDst,
                                           int lane, int a, int b) {
    int n = a;
    for (; n + 4 <= b; n += 4) {
        const float* g = src + (size_t)n * DIM;
        float*       l = ldsDst + n * DIM + lane * 4;
        __builtin_amdgcn_global_load_async_to_lds_b128((AS1 v4i*)g, (AS3 v4i*)l,    0, CPOL_NT);
        __builtin_amdgcn_global_load_async_to_lds_b128((AS1 v4i*)g, (AS3 v4i*)l,  512, CPOL_NT);
        __builtin_amdgcn_global_load_async_to_lds_b128((AS1 v4i*)g, (AS3 v4i*)l, 1024, CPOL_NT);
        __builtin_amdgcn_global_load_async_to_lds_b128((AS1 v4i*)g, (AS3 v4i*)l, 1536, CPOL_NT);
    }
    for (; n < b; ++n) {
        __builtin_amdgcn_global_load_async_to_lds_b128(
            (AS1 v4i*)(src + (size_t)n * DIM),
            (AS3 v4i*)(ldsDst + n * DIM + lane * 4), 0, CPOL_NT);
    }
}

__device__ __forceinline__ void reduce_rows(const float* __restrict__ lsrc,
                                            int lane, int a, int b, v4f& acc) {
    #pragma unroll 4
    for (int n = a; n < b; ++n) {
        v4f v = *(const v4f*)(lsrc + n * DIM + lane * 4);   // ds_load_b128
        acc.x += v.x; acc.y += v.y; acc.z += v.z; acc.w += v.w;
    }
}
#endif

__global__ __launch_bounds__(WAVES_PER_BLOCK * 32)
void mean_agg_kernel(const float* __restrict__ X,
                     const int* __restrict__ lengths,
                     float* __restrict__ out,
                     int P)
{
#if defined(HAVE_ASYNC_LDS)
    // 4 waves x 64 rows x 512B = 128 KiB static LDS (<< 320 KB WGP limit)
    __shared__ __align__(16) float lds[WAVES_PER_BLOCK * MAXN * DIM];
#endif

    const int lane = threadIdx.x & 31;
    const int wave = threadIdx.x >> 5;
    const int p    = blockIdx.x * WAVES_PER_BLOCK + wave;
    if (p >= P) return;                       // wave-uniform exit

    // Wave-uniform neighbor count -> scalar loop control, no divergence.
    int len = __builtin_amdgcn_readfirstlane(lengths[p]);

    // Lane l owns 16 contiguous bytes of each 512B row; one b128 per lane,
    // one full row per wave instruction -> perfectly coalesced.
    const float* src = X + (size_t)p * (MAXN * DIM) + lane * 4;

    v4f acc = {0.f, 0.f, 0.f, 0.f};

#if defined(HAVE_ASYNC_LDS)
    float* lwave = &lds[wave * (MAXN * DIM)];

    // Two-chunk pipeline with constant-immediate waits only:
    //   issue A -> wait 0 -> issue B (in flight) || reduce A -> wait 0 -> reduce B
    // len <= 63 keeps every phase within the 6-bit ASYNCcnt.
    const int nA = (len < 32) ? len : 32;

    issue_rows(src, lwave, lane, 0, nA);      // chunk A: up to 16 KB in flight
    wait_async_zero();                        // A landed in LDS
    issue_rows(src, lwave, lane, nA, len);    // chunk B streams while we reduce A
    reduce_rows(lwave, lane, 0, nA, acc);
    wait_async_zero();                        // B landed
    reduce_rows(lwave, lane, nA, len, acc);
#else
    // Fallback: direct streaming b128 loads.
    #pragma unroll 4
    for (int n = 0; n < len; ++n) {
        v4f v = __builtin_nontemporal_load((const v4f*)(src + n * DIM));
        acc.x += v.x; acc.y += v.y; acc.z += v.z; acc.w += v.w;
    }
#endif

    // Reference divides by the (constant) feature dim D = 128.
    const float inv = 1.0f / (float)DIM;
    acc.x *= inv; acc.y *= inv; acc.z *= inv; acc.w *= inv;

    // Streamed-once output: non-temporal b128 store.
    __builtin_nontemporal_store(acc, (v4f*)(out + (size_t)p * DIM + lane * 4));
}

extern "C" void kernel_launch(void* const* d_in, const int* in_sizes, int n_in,
                              void* d_out, int out_size, void* d_ws, size_t ws_size,
                              hipStream_t stream) {
    const float* X       = (const float*)d_in[0];
    const int*   lengths = (const int*)d_in[1];
    float*       out     = (float*)d_out;

    const int P = in_sizes[1];                 // B*V = 16384
    dim3 block(WAVES_PER_BLOCK * 32);
    dim3 grid((P + WAVES_PER_BLOCK - 1) / WAVES_PER_BLOCK);
    hipLaunchKernelGGL(mean_agg_kernel, grid, block, 0, stream, X, lengths, out, P);
}